// PrecisionLoss_20366734917885
// MI455X (gfx1250) — compile-verified
//
#include <hip/hip_runtime.h>
#include <math.h>

// ---------------- problem constants (from reference) ----------------
#define BATCH   32
#define PRI     24564
#define NCLS    21
#define NOBJ    20
#define TOPK    200
#define NROW    (BATCH * PRI)
#define NMS_TH  0.45f
#define CONF_TH 0.01f
#define OV_TH   0.5f

typedef __attribute__((ext_vector_type(2))) float v2f;
typedef __attribute__((ext_vector_type(8))) float v8f;

// -------------------------------------------------------------------
// 32-lane f32 sum via V_WMMA_F32_16X16X4_F32.
// Lane L's value lands in A[m, k]: lanes 0-15 -> (m=L, k=0),
// lanes 16-31 -> (m=L-16, k=2); A's second VGPR (k=1/3) is zero.
// B = all-ones so D[m,n] = val[m] + val[m+16] for every n.
// Each lane then sums its 8 D rows; __shfl_xor(16) pairs rows 0-7
// with rows 8-15, leaving the full 32-lane sum on every lane.
// Must be called with EXEC all-ones (wave-uniform branch).
// -------------------------------------------------------------------
__device__ __forceinline__ float wave_sum32_wmma(float v) {
  v2f a; a[0] = v;    a[1] = 0.0f;
  v2f b; b[0] = 1.0f; b[1] = 1.0f;
  v8f c = {0.f, 0.f, 0.f, 0.f, 0.f, 0.f, 0.f, 0.f};
  v8f d = __builtin_amdgcn_wmma_f32_16x16x4_f32(
      /*neg_a=*/false, a, /*neg_b=*/false, b,
      /*c_mod=*/(short)0, c, /*reuse_a=*/false, /*reuse_b=*/false);
  float s = d[0] + d[1] + d[2] + d[3] + d[4] + d[5] + d[6] + d[7];
  s += __shfl_xor(s, 16, 32);
  return s;
}

// -------------------------------------------------------------------
// K1: per-image prior<->truth matching + encode.  One block per image.
// -------------------------------------------------------------------
__global__ void k_match(const float* __restrict__ priors,
                        const float* __restrict__ targets,
                        float* __restrict__ loc_t, int* __restrict__ conf_t,
                        float* __restrict__ bto, int* __restrict__ bti) {
  const int b   = blockIdx.x;
  const int tid = threadIdx.x;
  const int NT  = 256;

  __shared__ float tr[NOBJ][4];
  __shared__ float tarea[NOBJ];
  __shared__ float tlabel[NOBJ];
  __shared__ float bestOvS[NOBJ][256];
  __shared__ int   bestIdS[NOBJ][256];
  __shared__ int   bestPrior[NOBJ];

  if (tid < NOBJ) {
    const float* t = targets + (size_t)(b * NOBJ + tid) * 5;
    tr[tid][0] = t[0]; tr[tid][1] = t[1]; tr[tid][2] = t[2]; tr[tid][3] = t[3];
    tlabel[tid] = t[4];
    tarea[tid]  = (t[2] - t[0]) * (t[3] - t[1]);
  }
  __syncthreads();

  float bOv[NOBJ];
  int   bId[NOBJ];
#pragma unroll
  for (int o = 0; o < NOBJ; ++o) { bOv[o] = -1.0f; bId[o] = 0x7fffffff; }

  for (int p = tid; p < PRI; p += NT) {
    const float* pr = priors + (size_t)p * 4;
    const float w = pr[2], h = pr[3];
    const float px0 = pr[0] - 0.5f * w, py0 = pr[1] - 0.5f * h;
    const float px1 = pr[0] + 0.5f * w, py1 = pr[1] + 0.5f * h;
    const float parea = w * h;
    float bo = -1.0f; int bidx = 0;
#pragma unroll
    for (int o = 0; o < NOBJ; ++o) {
      const float lx = fmaxf(tr[o][0], px0), ly = fmaxf(tr[o][1], py0);
      const float rx = fminf(tr[o][2], px1), ry = fminf(tr[o][3], py1);
      const float iw = fmaxf(rx - lx, 0.f), ih = fmaxf(ry - ly, 0.f);
      const float inter = iw * ih;
      const float iou = inter / (tarea[o] + parea - inter);
      if (iou > bo) { bo = iou; bidx = o; }                 // first-max over truths
      if (iou > bOv[o]) { bOv[o] = iou; bId[o] = p; }       // first-max over priors
    }
    bto[(size_t)b * PRI + p] = bo;
    bti[(size_t)b * PRI + p] = bidx;
  }
#pragma unroll
  for (int o = 0; o < NOBJ; ++o) { bestOvS[o][tid] = bOv[o]; bestIdS[o][tid] = bId[o]; }
  __syncthreads();

  if (tid < NOBJ) {                       // reduce best prior per truth
    float bv = -1.0f; int bi = 0x7fffffff;
    for (int t = 0; t < NT; ++t) {
      const float v = bestOvS[tid][t]; const int i = bestIdS[tid][t];
      if (v > bv || (v == bv && i < bi)) { bv = v; bi = i; }
    }
    bestPrior[tid] = bi;
  }
  __syncthreads();

  if (tid == 0) {                          // scatter overrides; last truth wins
    for (int o = 0; o < NOBJ; ++o) {
      const int p = bestPrior[o];
      bto[(size_t)b * PRI + p] = 2.0f;
      bti[(size_t)b * PRI + p] = o;
    }
  }
  __syncthreads();

  for (int p = tid; p < PRI; p += NT) {    // encode + conf target
    const float ov = bto[(size_t)b * PRI + p];
    const int   o  = bti[(size_t)b * PRI + p];
    const float mx0 = tr[o][0], my0 = tr[o][1], mx1 = tr[o][2], my1 = tr[o][3];
    const float* pr = priors + (size_t)p * 4;
    float* lt = loc_t + ((size_t)b * PRI + p) * 4;
    lt[0] = ((mx0 + mx1) * 0.5f - pr[0]) / (0.1f * pr[2]);
    lt[1] = ((my0 + my1) * 0.5f - pr[1]) / (0.1f * pr[3]);
    lt[2] = logf((mx1 - mx0) / pr[2]) * (1.0f / 0.2f);
    lt[3] = logf((my1 - my0) / pr[3]) * (1.0f / 0.2f);
    const int cf = (int)tlabel[o] + 1;
    conf_t[(size_t)b * PRI + p] = (ov < OV_TH) ? 0 : cf;
  }
}

// -------------------------------------------------------------------
// K2: softmax over classes + box decode + clear kept mask. Thread/row.
// -------------------------------------------------------------------
__global__ void k_softmax_decode(const float* __restrict__ conf,
                                 const float* __restrict__ loc,
                                 const float* __restrict__ priors,
                                 float* __restrict__ probs,
                                 float* __restrict__ dec,
                                 int* __restrict__ kept) {
  const int r = blockIdx.x * blockDim.x + threadIdx.x;
  if (r >= NROW) return;
  const float* x = conf + (size_t)r * NCLS;
  float m = x[0];
#pragma unroll
  for (int c = 1; c < NCLS; ++c) m = fmaxf(m, x[c]);
  float e[NCLS]; float s = 0.f;
#pragma unroll
  for (int c = 0; c < NCLS; ++c) { e[c] = __expf(x[c] - m); s += e[c]; }
  const float inv = 1.0f / s;
  float* pr = probs + (size_t)r * NCLS;
#pragma unroll
  for (int c = 0; c < NCLS; ++c) pr[c] = e[c] * inv;

  const int p = r % PRI;
  const float* pb = priors + (size_t)p * 4;
  const float* lo = loc + (size_t)r * 4;
  const float cx = pb[0] + lo[0] * 0.1f * pb[2];
  const float cy = pb[1] + lo[1] * 0.1f * pb[3];
  const float w  = pb[2] * __expf(lo[2] * 0.2f);
  const float h  = pb[3] * __expf(lo[3] * 0.2f);
  float* d = dec + (size_t)r * 4;
  d[0] = cx - 0.5f * w; d[1] = cy - 0.5f * h;
  d[2] = cx + 0.5f * w; d[3] = cy + 0.5f * h;
  kept[r] = 0;
}

// -------------------------------------------------------------------
// K3: per-(image,class) top-200 + greedy NMS. One block per chain.
// Scores staged in LDS (98 KB of the 320 KB WGP pool).
// -------------------------------------------------------------------
#define TAKEN_W ((PRI + 31) / 32)
__global__ void k_nms(const float* __restrict__ probs,
                      const float* __restrict__ dec,
                      int* __restrict__ kept) {
  extern __shared__ char smem[];
  float*    sc     = (float*)smem;                 // PRI
  unsigned* taken  = (unsigned*)(sc + PRI);        // TAKEN_W
  float*    rv     = (float*)(taken + TAKEN_W);    // 256
  int*      ri     = (int*)(rv + 256);             // 256
  float*    bx     = (float*)(ri + 256);           // TOPK*4
  float*    tops   = (float*)(bx + TOPK * 4);      // TOPK
  int*      topi   = (int*)(tops + TOPK);          // TOPK
  int*      validk = (int*)(topi + TOPK);          // TOPK
  int*      keepfl = (int*)(validk + TOPK);        // TOPK
  int*      supp   = (int*)(keepfl + TOPK);        // 1

  const int NT  = 256;
  const int tid = threadIdx.x;
  const int b   = blockIdx.x / (NCLS - 1);
  const int c   = (blockIdx.x % (NCLS - 1)) + 1;
  const float NEG_INF = -__builtin_inff();

  for (int p = tid; p < PRI; p += NT) {
    const float v = probs[((size_t)b * PRI + p) * NCLS + c];
    sc[p] = (v > CONF_TH) ? v : NEG_INF;
  }
  for (int i = tid; i < TAKEN_W; i += NT) taken[i] = 0u;
  if (tid == 0) *supp = 0;
  __syncthreads();

  // ---- iterative top-200 (ties -> lowest index, like lax.top_k) ----
  for (int k = 0; k < TOPK; ++k) {
    float bv = NEG_INF; int bi = 0x7fffffff;
    for (int p = tid; p < PRI; p += NT) {
      if ((taken[p >> 5] >> (p & 31)) & 1u) continue;
      const float v = sc[p];
      if (v > bv || (v == bv && p < bi)) { bv = v; bi = p; }
    }
    rv[tid] = bv; ri[tid] = bi;
    __syncthreads();
    for (int off = NT / 2; off > 0; off >>= 1) {
      if (tid < off) {
        const float v2 = rv[tid + off]; const int i2 = ri[tid + off];
        if (v2 > rv[tid] || (v2 == rv[tid] && i2 < ri[tid])) { rv[tid] = v2; ri[tid] = i2; }
      }
      __syncthreads();
    }
    if (tid == 0) {
      const int sel = ri[0];
      topi[k] = sel; tops[k] = rv[0];
      taken[sel >> 5] |= 1u << (sel & 31);
    }
    __syncthreads();
  }

  // ---- stage candidate boxes ----
  for (int k = tid; k < TOPK; k += NT) {
    const float* d = dec + ((size_t)b * PRI + topi[k]) * 4;
    bx[k * 4 + 0] = d[0]; bx[k * 4 + 1] = d[1];
    bx[k * 4 + 2] = d[2]; bx[k * 4 + 3] = d[3];
    validk[k] = (tops[k] > CONF_TH) ? 1 : 0;
    keepfl[k] = 0;
  }
  __syncthreads();

  float mx0 = 0.f, my0 = 0.f, mx1 = 0.f, my1 = 0.f, marea = 0.f;
  if (tid < TOPK) {
    mx0 = bx[tid * 4 + 0]; my0 = bx[tid * 4 + 1];
    mx1 = bx[tid * 4 + 2]; my1 = bx[tid * 4 + 3];
    marea = (mx1 - mx0) * (my1 - my0);
  }

  // ---- greedy suppression: serial over i, parallel over earlier j ----
  for (int i = 0; i < TOPK; ++i) {
    if (tid < i && keepfl[tid]) {
      const float ix0 = bx[i * 4 + 0], iy0 = bx[i * 4 + 1];
      const float ix1 = bx[i * 4 + 2], iy1 = bx[i * 4 + 3];
      const float lx = fmaxf(ix0, mx0), ly = fmaxf(iy0, my0);
      const float rx = fminf(ix1, mx1), ry = fminf(iy1, my1);
      const float iw = fmaxf(rx - lx, 0.f), ih = fmaxf(ry - ly, 0.f);
      const float inter = iw * ih;
      const float iarea = (ix1 - ix0) * (iy1 - iy0);
      const float iou = inter / (iarea + marea - inter);
      if (iou > NMS_TH) *supp = 1;     // benign write race (same value)
    }
    __syncthreads();
    if (tid == i) { keepfl[i] = (validk[i] && (*supp == 0)) ? 1 : 0; *supp = 0; }
    __syncthreads();
  }

  if (tid < TOPK && keepfl[tid])
    atomicOr(&kept[(size_t)b * PRI + topi[tid]], 1);
}

// -------------------------------------------------------------------
// K4: per-row losses + WMMA block reduce -> per-block partials.
// -------------------------------------------------------------------
__global__ void k_loss(const float* __restrict__ probs,
                       const float* __restrict__ loc,
                       const float* __restrict__ loc_t,
                       const int* __restrict__ conf_t,
                       const int* __restrict__ kept,
                       float* __restrict__ partials) {
  float ll = 0.f, lc = 0.f, ne = 0.f;
  for (int r = blockIdx.x * blockDim.x + threadIdx.x; r < NROW;
       r += gridDim.x * blockDim.x) {
    const float* pr = probs + (size_t)r * NCLS;
    float cp[NCLS];
    cp[0] = 0.f;
    bool eff = false;
#pragma unroll
    for (int c = 1; c < NCLS; ++c) {
      const float v = pr[c];
      cp[c] = (v > CONF_TH) ? v : 0.f;
      eff = eff || (cp[c] != 0.f);
    }
    if (!eff) continue;
    float m = 0.f;
#pragma unroll
    for (int c = 1; c < NCLS; ++c) m = fmaxf(m, cp[c]);
    float s = 0.f;
#pragma unroll
    for (int c = 0; c < NCLS; ++c) s += __expf(cp[c] - m);
    const float lse = logf(s) + m;
    const int t = conf_t[r];
    const float pt = probs[(size_t)r * NCLS + t];
    const float cpt = (t == 0) ? 0.f : ((pt > CONF_TH) ? pt : 0.f);
    lc += lse - cpt;
    ne += 1.f;
    const int ka = kept[r];
#pragma unroll
    for (int i = 0; i < 4; ++i) {
      const float lp = ka ? loc[(size_t)r * 4 + i] : 0.f;
      const float d = fabsf(lp - loc_t[(size_t)r * 4 + i]);
      ll += (d < 1.f) ? 0.5f * d * d : d - 0.5f;
    }
  }

  __shared__ float red[3][256];
  red[0][threadIdx.x] = ll; red[1][threadIdx.x] = lc; red[2][threadIdx.x] = ne;
  __syncthreads();
  if (threadIdx.x < 32) {                 // wave-uniform: wave0, EXEC all-ones
    float a0 = 0.f, a1 = 0.f, a2 = 0.f;
    for (int k = threadIdx.x; k < 256; k += 32) {
      a0 += red[0][k]; a1 += red[1][k]; a2 += red[2][k];
    }
    a0 = wave_sum32_wmma(a0);
    a1 = wave_sum32_wmma(a1);
    a2 = wave_sum32_wmma(a2);
    if (threadIdx.x == 0) {
      partials[(size_t)blockIdx.x * 3 + 0] = a0;
      partials[(size_t)blockIdx.x * 3 + 1] = a1;
      partials[(size_t)blockIdx.x * 3 + 2] = a2;
    }
  }
}

// -------------------------------------------------------------------
// K5: final reduce (WMMA) + normalization.  out = (loss_l/N, loss_c/N)
// -------------------------------------------------------------------
__global__ void k_final(const float* __restrict__ partials, int nblk,
                        float* __restrict__ out) {
  float a0 = 0.f, a1 = 0.f, a2 = 0.f;
  for (int i = threadIdx.x; i < nblk; i += 256) {
    a0 += partials[(size_t)i * 3 + 0];
    a1 += partials[(size_t)i * 3 + 1];
    a2 += partials[(size_t)i * 3 + 2];
  }
  __shared__ float red[3][256];
  red[0][threadIdx.x] = a0; red[1][threadIdx.x] = a1; red[2][threadIdx.x] = a2;
  __syncthreads();
  if (threadIdx.x < 32) {
    float s0 = 0.f, s1 = 0.f, s2 = 0.f;
    for (int k = threadIdx.x; k < 256; k += 32) {
      s0 += red[0][k]; s1 += red[1][k]; s2 += red[2][k];
    }
    s0 = wave_sum32_wmma(s0);
    s1 = wave_sum32_wmma(s1);
    s2 = wave_sum32_wmma(s2);
    if (threadIdx.x == 0) {
      const float N = fmaxf(s2 * (float)NCLS, 1.0f);
      out[0] = s0 / N;   // loss_l / N
      out[1] = s1 / N;   // loss_c / N
    }
  }
}

// -------------------------------------------------------------------
extern "C" void kernel_launch(void* const* d_in, const int* in_sizes, int n_in,
                              void* d_out, int out_size, void* d_ws, size_t ws_size,
                              hipStream_t stream) {
  const float* loc_data  = (const float*)d_in[0];
  const float* conf_data = (const float*)d_in[1];
  const float* priors    = (const float*)d_in[2];
  const float* targets   = (const float*)d_in[3];
  float* out = (float*)d_out;

  // workspace partition (all L2-resident: ~104 MB of the 192 MB L2)
  char* w = (char*)d_ws;
  float* probs  = (float*)w; w += (size_t)NROW * NCLS * sizeof(float);
  float* dec    = (float*)w; w += (size_t)NROW * 4 * sizeof(float);
  float* loc_t  = (float*)w; w += (size_t)NROW * 4 * sizeof(float);
  int*   conf_t = (int*)w;   w += (size_t)NROW * sizeof(int);
  int*   kept   = (int*)w;   w += (size_t)NROW * sizeof(int);
  float* bto    = (float*)w; w += (size_t)NROW * sizeof(float);
  int*   bti    = (int*)w;   w += (size_t)NROW * sizeof(int);
  float* partials = (float*)w;

  const int LOSS_BLOCKS = 512;
  const size_t NMS_SMEM =
      ((size_t)PRI + TAKEN_W + 256 + 256 + TOPK * 4 + TOPK * 4 + 1) * 4;

  k_match<<<BATCH, 256, 0, stream>>>(priors, targets, loc_t, conf_t, bto, bti);
  k_softmax_decode<<<(NROW + 255) / 256, 256, 0, stream>>>(
      conf_data, loc_data, priors, probs, dec, kept);
  k_nms<<<BATCH * (NCLS - 1), 256, NMS_SMEM, stream>>>(probs, dec, kept);
  k_loss<<<LOSS_BLOCKS, 256, 0, stream>>>(probs, loc_data, loc_t, conf_t, kept,
                                          partials);
  k_final<<<1, 256, 0, stream>>>(partials, LOSS_BLOCKS, out);
}